// SplitConvStack_66383014527669
// MI455X (gfx1250) — compile-verified
//
#include <hip/hip_runtime.h>
#include <cstdint>

#define GAS __attribute__((address_space(1)))
#define LAS __attribute__((address_space(3)))

typedef int v4i_t __attribute__((__vector_size__(16)));
typedef GAS v4i_t* gv4i_p;   // global (AS1) int4*
typedef LAS v4i_t* lv4i_p;   // LDS (AS3) int4*

#ifndef __has_builtin
#define __has_builtin(x) 0
#endif

#if __has_builtin(__builtin_amdgcn_global_load_async_to_lds_b128)
#define USE_ASYNC 1
#else
#define USE_ASYNC 0
#endif

static constexpr int Wd   = 4096;
static constexpr int Cd   = 64;
static constexpr int TILE = 256;
static constexpr int NIN  = 305;  // input rows needed: TILE + 24 left + 25 right
static constexpr int NINP = 306;  // +1 pad row (sliding-window over-read)
static constexpr int N1   = 298;  // stage1 rows: TILE + 21 + 21
static constexpr int N1P  = 300;  // +2 pad rows
static constexpr int N2   = 284;  // stage2 rows: TILE + 14 + 14

__device__ __forceinline__ float2 fma2(float2 a, float2 b, float2 c) {
  float2 r; r.x = fmaf(a.x, b.x, c.x); r.y = fmaf(a.y, b.y, c.y); return r;
}
__device__ __forceinline__ int imin(int a, int b) { return a < b ? a : b; }

__global__ __launch_bounds__(256)
void dwconv3_fused_kernel(const float* __restrict__ x,
                          const float* __restrict__ w1,
                          const float* __restrict__ w2,
                          const float* __restrict__ w3,
                          float* __restrict__ out) {
  extern __shared__ float smem[];
  float*  sA  = smem;              // NINP * 64 floats (input, then stage2 result)
  float*  sB  = smem + NINP * Cd;  // N1P  * 64 floats (stage1 result)
  float2* sA2 = (float2*)sA;
  float2* sB2 = (float2*)sB;

  const int tid  = threadIdx.x;
  const int lane = tid & 31;       // channel pair: channels {2*lane, 2*lane+1}
  const int g    = tid >> 5;       // position group 0..7
  const int w0   = blockIdx.x * TILE;
  const int bh   = blockIdx.y;     // b*H + h, 0..335
  const size_t rowBase = (size_t)bh * Wd * Cd;

  // ---- Stage 0: global -> LDS input tile (zero-filled SAME halo) ----
  // 305 positions * 16 float4 (64 channels) = 4880 float4 chunks
  for (int e = tid; e < NIN * 16; e += 256) {
    const int pos = e >> 4;
    const int c4  = (e & 15) << 2;
    const int w   = w0 - 24 + pos;
    float* lp = &sA[pos * Cd + c4];
    if ((unsigned)w < (unsigned)Wd) {
      const float* gp = x + rowBase + (size_t)w * Cd + c4;
#if USE_ASYNC
      __builtin_amdgcn_global_load_async_to_lds_b128(
          (gv4i_p)(uintptr_t)gp,
          (lv4i_p)(uint32_t)(uintptr_t)lp,
          0, 0);
#else
      *(float4*)lp = *(const float4*)gp;
#endif
    } else {
      float4 z = {0.f, 0.f, 0.f, 0.f};
      *(float4*)lp = z;
    }
  }
#if USE_ASYNC
#if __has_builtin(__builtin_amdgcn_s_wait_asynccnt)
  __builtin_amdgcn_s_wait_asynccnt(0);
#else
  asm volatile("s_wait_asynccnt 0" ::: "memory");
#endif
#endif
  __syncthreads();

  // ---- Stage 1: dilation 1, pad_lo=3. s1[j] (w = w0-21+j) = relu(sum_k x[j+k]*w1[k]) ----
  {
    float2 wt[8];
#pragma unroll
    for (int k = 0; k < 8; ++k) wt[k] = *(const float2*)(w1 + k * Cd + 2 * lane);
    const int jb = g * 38;
    const int je = imin(jb + 38, N1);
    if (jb < je) {
      float2 win[8];
#pragma unroll
      for (int k = 0; k < 8; ++k) win[k] = sA2[(jb + k) * 32 + lane];
      for (int j = jb; j < je; ++j) {
        float2 acc = {0.f, 0.f};
#pragma unroll
        for (int k = 0; k < 8; ++k) acc = fma2(win[k], wt[k], acc);
        const int w = w0 - 21 + j;
        const bool ok = (unsigned)w < (unsigned)Wd;   // intermediate SAME padding
        float2 r;
        r.x = ok ? fmaxf(acc.x, 0.f) : 0.f;
        r.y = ok ? fmaxf(acc.y, 0.f) : 0.f;
        sB2[j * 32 + lane] = r;
#pragma unroll
        for (int k = 0; k < 7; ++k) win[k] = win[k + 1];
        win[7] = sA2[(j + 8) * 32 + lane];            // row <= 305 (padded)
      }
    }
  }
  __syncthreads();

  // ---- Stage 2: dilation 2, pad_lo=7. s2[j] (w = w0-14+j) = relu(sum_k s1[j+2k]*w2[k]) ----
  {
    float2 wt[8];
#pragma unroll
    for (int k = 0; k < 8; ++k) wt[k] = *(const float2*)(w2 + k * Cd + 2 * lane);
    const int jb = g * 36;
    const int je = imin(jb + 36, N2);
    for (int ph = 0; ph < 2; ++ph) {                  // dense per parity phase
      const int j0 = jb + ph;
      if (j0 >= je) continue;
      float2 win[8];
#pragma unroll
      for (int k = 0; k < 8; ++k) win[k] = sB2[(j0 + 2 * k) * 32 + lane];
      for (int j = j0; j < je; j += 2) {
        float2 acc = {0.f, 0.f};
#pragma unroll
        for (int k = 0; k < 8; ++k) acc = fma2(win[k], wt[k], acc);
        const int w = w0 - 14 + j;
        const bool ok = (unsigned)w < (unsigned)Wd;
        float2 r;
        r.x = ok ? fmaxf(acc.x, 0.f) : 0.f;
        r.y = ok ? fmaxf(acc.y, 0.f) : 0.f;
        sA2[j * 32 + lane] = r;                       // reuse input buffer
#pragma unroll
        for (int k = 0; k < 7; ++k) win[k] = win[k + 1];
        win[7] = sB2[(j + 16) * 32 + lane];           // row <= 299 (padded)
      }
    }
  }
  __syncthreads();

  // ---- Stage 3: dilation 4, pad_lo=14. y[j] (w = w0+j) = relu(sum_k s2[j+4k]*w3[k]) ----
  {
    float2 wt[8];
#pragma unroll
    for (int k = 0; k < 8; ++k) wt[k] = *(const float2*)(w3 + k * Cd + 2 * lane);
    const int jb = g * 32;
    float2* out2 = (float2*)out;
    for (int ph = 0; ph < 4; ++ph) {                  // dense per mod-4 phase
      const int j0 = jb + ph;
      float2 win[8];
#pragma unroll
      for (int k = 0; k < 8; ++k) win[k] = sA2[(j0 + 4 * k) * 32 + lane];
      for (int j = j0; j < jb + 32; j += 4) {
        float2 acc = {0.f, 0.f};
#pragma unroll
        for (int k = 0; k < 8; ++k) acc = fma2(win[k], wt[k], acc);
        float2 r;
        r.x = fmaxf(acc.x, 0.f);
        r.y = fmaxf(acc.y, 0.f);
        out2[((size_t)bh * Wd + w0 + j) * 32 + lane] = r;
#pragma unroll
        for (int k = 0; k < 7; ++k) win[k] = win[k + 1];
        win[7] = sA2[(j + 32) * 32 + lane];           // row <= 287 < 306
      }
    }
  }
}

extern "C" void kernel_launch(void* const* d_in, const int* in_sizes, int n_in,
                              void* d_out, int out_size, void* d_ws, size_t ws_size,
                              hipStream_t stream) {
  const float* x  = (const float*)d_in[0];
  const float* w1 = (const float*)d_in[1];
  const float* w2 = (const float*)d_in[2];
  const float* w3 = (const float*)d_in[3];
  float* out = (float*)d_out;

  dim3 grid(Wd / TILE, 16 * 21);   // 16 W-tiles x (B*H = 336) rows
  dim3 block(256);
  size_t shmem = (size_t)(NINP + N1P) * Cd * sizeof(float);  // 155,136 B
  hipLaunchKernelGGL(dwconv3_fused_kernel, grid, block, shmem, stream,
                     x, w1, w2, w3, out);
}